// GraphEncoder_32581621907908
// MI455X (gfx1250) — compile-verified
//
#include <hip/hip_runtime.h>
#include <math.h>

#define B_ 4
#define N_ 1024
#define NP_ 1025
#define E_ 16384
#define D_ 256
#define H_ 8
#define DH_ 32
#define EPS_ 1e-6f
#define NEG_ -1.0e9f
#define MROWS (B_ * NP_)   // 4100

typedef __attribute__((ext_vector_type(2))) float v2f;
typedef __attribute__((ext_vector_type(4))) float v4f;
typedef __attribute__((ext_vector_type(8))) float v8f;

// ---------------- device helpers ----------------

__device__ __forceinline__ float wsum32(float v) {
  v += __shfl_xor(v, 1, 32);
  v += __shfl_xor(v, 2, 32);
  v += __shfl_xor(v, 4, 32);
  v += __shfl_xor(v, 8, 32);
  v += __shfl_xor(v, 16, 32);
  return v;
}

// monotone float<->uint encoding for atomic max on floats
__device__ __forceinline__ unsigned fenc(float f) {
  unsigned u = __float_as_uint(f);
  return (u >> 31) ? ~u : (u | 0x80000000u);
}
__device__ __forceinline__ float fdec(unsigned u) {
  return __uint_as_float((u >> 31) ? (u & 0x7FFFFFFFu) : ~u);
}

// ---------------- WMMA GEMM: C[M,Nc] = act(A[M,K] @ W[K,Nc] + bias) ----------------
// Requirements: K % 8 == 0, Nc % 256 == 0 (all launches here use Nc=256).
// Block = 256 threads = 8 waves; wave tile 16x64; block tile 32x256.
__global__ void gemm_wmma(const float* __restrict__ A, const float* __restrict__ W,
                          const float* __restrict__ bias, float* __restrict__ C,
                          int M, int K, int Nc, int act) {
  const int wave  = threadIdx.x >> 5;
  const int lane  = threadIdx.x & 31;
  const int waveM = wave >> 2;       // 0..1
  const int waveN = wave & 3;        // 0..3
  const int rowBase = blockIdx.x * 32 + waveM * 16;
  const int colBase = blockIdx.y * 256 + waveN * 64;
  const int laneHalf = lane & 15;
  const int laneHi   = lane >> 4;    // 0: lanes 0-15, 1: lanes 16-31

  const int aRow = rowBase + laneHalf;
  const bool aValid = aRow < M;
  // Clamp OOB lanes to row 0: their accumulator rows are never stored, so the
  // garbage contribution is harmless and the inner loop stays branch-free.
  const float* __restrict__ Ap = A + (size_t)(aValid ? aRow : 0) * K;

  v8f acc[4];
#pragma unroll
  for (int t = 0; t < 4; ++t) acc[t] = (v8f){0.f,0.f,0.f,0.f,0.f,0.f,0.f,0.f};

  for (int k = 0; k < K; k += 8) {
    const int kA = k + laneHi * 2;
    const v2f a0 = *(const v2f*)(Ap + kA);
    const v2f a1 = *(const v2f*)(Ap + kA + 4);
#pragma unroll
    for (int t = 0; t < 4; ++t) {
      const int col = colBase + t * 16 + laneHalf;
      v2f b0, b1;
      b0.x = W[(size_t)kA * Nc + col];
      b0.y = W[(size_t)(kA + 1) * Nc + col];
      b1.x = W[(size_t)(kA + 4) * Nc + col];
      b1.y = W[(size_t)(kA + 5) * Nc + col];
      acc[t] = __builtin_amdgcn_wmma_f32_16x16x4_f32(
          false, a0, false, b0, (short)0, acc[t], false, false);
      acc[t] = __builtin_amdgcn_wmma_f32_16x16x4_f32(
          false, a1, false, b1, (short)0, acc[t], false, false);
    }
  }

#pragma unroll
  for (int t = 0; t < 4; ++t) {
    const int col = colBase + t * 16 + laneHalf;
    const float bv = bias ? bias[col] : 0.f;
#pragma unroll
    for (int i = 0; i < 8; ++i) {
      const int row = rowBase + i + laneHi * 8;
      if (row < M) {
        float v = acc[t][i] + bv;
        if (act == 1 && v < 0.f) v = 0.f;
        C[(size_t)row * Nc + col] = v;
      }
    }
  }
}

// ---------------- row LayerNorm over D=256 (scale+bias, optional relu) ----------------
__global__ void ln_rows(const float* __restrict__ X, const float* __restrict__ s,
                        const float* __restrict__ bvec, float* __restrict__ Y, int relu) {
  __shared__ float red[256];
  const int row = blockIdx.x, d = threadIdx.x;
  float v = X[(size_t)row * D_ + d];
  red[d] = v; __syncthreads();
  for (int t = 128; t > 0; t >>= 1) { if (d < t) red[d] += red[d + t]; __syncthreads(); }
  const float mu = red[0] * (1.f / D_); __syncthreads();
  const float df = v - mu;
  red[d] = df * df; __syncthreads();
  for (int t = 128; t > 0; t >>= 1) { if (d < t) red[d] += red[d + t]; __syncthreads(); }
  const float var = red[0] * (1.f / D_);
  float y = df * rsqrtf(var + EPS_) * s[d] + bvec[d];
  if (relu && y < 0.f) y = 0.f;
  Y[(size_t)row * D_ + d] = y;
}

// ---------------- node feature prep: nf = concat(node_features, global) + zero pad row ----
__global__ void prep_nf(const float* __restrict__ nfeat, const float* __restrict__ gfeat,
                        float* __restrict__ nf) {
  const int i = blockIdx.x * 256 + threadIdx.x;
  const int tot = MROWS * 48;
  if (i >= tot) return;
  const int j = i % 48;
  const int n = (i / 48) % NP_;
  const int b = i / (48 * NP_);
  float v;
  if (n == N_)      v = 0.f;
  else if (j < 32)  v = nfeat[((size_t)(b * N_ + n)) * 32 + j];
  else              v = gfeat[b * 16 + (j - 32)];
  nf[i] = v;
}

__global__ void init_g(const float* __restrict__ glob, float* __restrict__ g) {
  const int i = blockIdx.x * 256 + threadIdx.x;
  if (i < B_ * D_) g[i] = glob[i % D_];
}

__global__ void concat_xg(const float* __restrict__ x, const float* __restrict__ g,
                          float* __restrict__ xc) {
  const int i = blockIdx.x * 256 + threadIdx.x;
  const int tot = MROWS * 512;
  if (i >= tot) return;
  const int j = i % 512;
  const int n = (i / 512) % NP_;
  const int b = i / (512 * NP_);
  xc[i] = (j < 256) ? x[((size_t)(b * NP_ + n)) * D_ + j] : g[b * D_ + (j - 256)];
}

// ---------------- attention: q projection + per-head scale-only LN (one wave == one head) --
__global__ void qproj_ln(const float* __restrict__ g, const float* __restrict__ Wq,
                         const float* __restrict__ bq, const float* __restrict__ lnq,
                         float* __restrict__ qln) {
  __shared__ float sg[256];
  const int b = blockIdx.x, d = threadIdx.x;
  sg[d] = g[b * D_ + d];
  __syncthreads();
  float acc = bq[d];
  for (int k = 0; k < D_; ++k) acc += sg[k] * Wq[(size_t)k * D_ + d];
  // per-head LN: head == wave (32 lanes)
  const float mu = wsum32(acc) * (1.f / 32.f);
  const float df = acc - mu;
  const float var = wsum32(df * df) * (1.f / 32.f);
  qln[b * D_ + d] = df * rsqrtf(var + EPS_) * lnq[d & 31];
}

// scores: one wave per (b,k,h); fused per-head LN of kh row
__global__ void attn_scores(const float* __restrict__ kh, const float* __restrict__ qln,
                            const float* __restrict__ lnk, const float* __restrict__ nmask,
                            float* __restrict__ scores) {
  const int wid = blockIdx.x * 8 + (threadIdx.x >> 5);
  const int lane = threadIdx.x & 31;
  if (wid >= B_ * NP_ * H_) return;
  const int h = wid % H_;
  const int k = (wid / H_) % NP_;
  const int b = wid / (H_ * NP_);
  const float v = kh[((size_t)(b * NP_ + k)) * D_ + h * DH_ + lane];
  const float mu = wsum32(v) * (1.f / 32.f);
  const float df = v - mu;
  const float var = wsum32(df * df) * (1.f / 32.f);
  const float kl = df * rsqrtf(var + EPS_) * lnk[lane];
  const float p = kl * qln[b * D_ + h * DH_ + lane];
  const float sc = wsum32(p) * 0.17677669529663687f;  // 1/sqrt(32)
  if (lane == 0) {
    const float mk = (k < N_) ? nmask[b * N_ + k] : 0.f;
    scores[(size_t)(b * H_ + h) * NP_ + k] = (mk > 0.f) ? sc : NEG_;
  }
}

// softmax over k and weighted sum of vh -> oh[b, h*32+d]; one block per (b,h)
__global__ void attn_softmax_o(const float* __restrict__ scores, const float* __restrict__ vh,
                               float* __restrict__ oh) {
  __shared__ float pr[NP_];
  __shared__ float red[256];
  __shared__ float part[8][DH_ + 1];
  const int b = blockIdx.x / H_, h = blockIdx.x % H_;
  const int tid = threadIdx.x;
  const float* sc = scores + (size_t)(b * H_ + h) * NP_;
  float lm = -3.4e38f;
  for (int k = tid; k < NP_; k += 256) lm = fmaxf(lm, sc[k]);
  red[tid] = lm; __syncthreads();
  for (int t = 128; t > 0; t >>= 1) { if (tid < t) red[tid] = fmaxf(red[tid], red[tid + t]); __syncthreads(); }
  const float mx = red[0]; __syncthreads();
  float ls = 0.f;
  for (int k = tid; k < NP_; k += 256) { float e = __expf(sc[k] - mx); pr[k] = e; ls += e; }
  red[tid] = ls; __syncthreads();
  for (int t = 128; t > 0; t >>= 1) { if (tid < t) red[tid] += red[tid + t]; __syncthreads(); }
  const float inv = 1.f / red[0]; __syncthreads();
  const int d = tid & 31, sub = tid >> 5;
  float acc = 0.f;
  for (int k = sub; k < NP_; k += 8)
    acc += pr[k] * vh[((size_t)(b * NP_ + k)) * D_ + h * DH_ + d];
  part[sub][d] = acc; __syncthreads();
  if (sub == 0) {
    float t = 0.f;
#pragma unroll
    for (int i = 0; i < 8; ++i) t += part[i][d];
    oh[b * D_ + h * DH_ + d] = t * inv;
  }
}

// out projection + residual + LN over 256; updates g in place (one block per b)
__global__ void attn_out_ln(const float* __restrict__ oh, const float* __restrict__ Wo,
                            const float* __restrict__ bo, const float* __restrict__ ln_s,
                            const float* __restrict__ ln_b, float* __restrict__ g) {
  __shared__ float so[256];
  __shared__ float red[256];
  const int b = blockIdx.x, d = threadIdx.x;
  so[d] = oh[b * D_ + d];
  const float gold = g[b * D_ + d];
  __syncthreads();
  float acc = bo[d];
  for (int k = 0; k < D_; ++k) acc += so[k] * Wo[(size_t)k * D_ + d];
  const float t = gold + acc;
  red[d] = t; __syncthreads();
  for (int s = 128; s > 0; s >>= 1) { if (d < s) red[d] += red[d + s]; __syncthreads(); }
  const float mu = red[0] * (1.f / D_); __syncthreads();
  const float df = t - mu;
  red[d] = df * df; __syncthreads();
  for (int s = 128; s > 0; s >>= 1) { if (d < s) red[d] += red[d + s]; __syncthreads(); }
  const float var = red[0] * (1.f / D_);
  g[b * D_ + d] = df * rsqrtf(var + EPS_) * ln_s[d] + ln_b[d];
}

// final: out = relu(LN(g))
__global__ void final_out(const float* __restrict__ g, const float* __restrict__ ln_s,
                          const float* __restrict__ ln_b, float* __restrict__ out) {
  __shared__ float red[256];
  const int b = blockIdx.x, d = threadIdx.x;
  const float t = g[b * D_ + d];
  red[d] = t; __syncthreads();
  for (int s = 128; s > 0; s >>= 1) { if (d < s) red[d] += red[d + s]; __syncthreads(); }
  const float mu = red[0] * (1.f / D_); __syncthreads();
  const float df = t - mu;
  red[d] = df * df; __syncthreads();
  for (int s = 128; s > 0; s >>= 1) { if (d < s) red[d] += red[d + s]; __syncthreads(); }
  const float var = red[0] * (1.f / D_);
  float y = df * rsqrtf(var + EPS_) * ln_s[d] + ln_b[d];
  out[b * D_ + d] = y > 0.f ? y : 0.f;
}

// ---------------- GAT edge kernels ----------------
__global__ void gat_init(unsigned* __restrict__ mmax, float* __restrict__ den,
                         float* __restrict__ outagg) {
  const int i = blockIdx.x * 256 + threadIdx.x;
  if (i < MROWS * D_) outagg[i] = 0.f;
  if (i < MROWS * H_) { mmax[i] = 0x007FFFFFu /* fenc(-inf) */; den[i] = 0.f; }
}

// one wave per edge: compute 8 head logits (fused ef@We in LDS) + atomic segment max
__global__ void gat_logits(const float* __restrict__ gs, const float* __restrict__ gr,
                           const float* __restrict__ ef, const int* __restrict__ el,
                           const unsigned char* __restrict__ emk,
                           const float* __restrict__ We, const float* __restrict__ be,
                           const float* __restrict__ attnW,
                           float* __restrict__ logits, unsigned* __restrict__ mmax) {
  __shared__ float sWe[16 * D_];   // 16 KB
  __shared__ float sBe[D_];
  __shared__ float sAt[H_ * DH_];
  const int tid = threadIdx.x;

  // Async bulk copy of We (16 KB) into LDS via the gfx1250 ASYNCcnt path.
  // IOFFSET is added to BOTH the LDS and global addresses (ISA 08 §4.4), so a
  // single base pair + 4 instruction offsets covers each thread's 64 bytes.
  {
    const unsigned lds_base =
        (unsigned)(uintptr_t)(&sWe[0]) + (unsigned)tid * 16u;
    const unsigned long long g_base =
        (unsigned long long)(uintptr_t)We + (unsigned long long)tid * 16ull;
    asm volatile(
        "global_load_async_to_lds_b128 %0, %1, off\n\t"
        "global_load_async_to_lds_b128 %0, %1, off offset:4096\n\t"
        "global_load_async_to_lds_b128 %0, %1, off offset:8192\n\t"
        "global_load_async_to_lds_b128 %0, %1, off offset:12288"
        :: "v"(lds_base), "v"(g_base) : "memory");
  }
  if (tid < D_) sBe[tid] = be[tid];
  if (tid < H_ * DH_) sAt[tid] = attnW[tid];

  // Edge metadata loads overlap with the async LDS fill.
  const int wave = tid >> 5, lane = tid & 31;
  const int eg = blockIdx.x * 8 + wave;
  const int egc = eg < B_ * E_ ? eg : 0;
  const int b = egc / E_;
  const bool mk = emk[egc] != 0;
  const int s = mk ? el[(size_t)egc * 2] : N_;
  const int r = mk ? el[(size_t)egc * 2 + 1] : N_;
  v4f ef4[4];
#pragma unroll
  for (int q = 0; q < 4; ++q) ef4[q] = *(const v4f*)(ef + (size_t)egc * 16 + q * 4);
  const float* gsp = gs + ((size_t)(b * NP_ + s)) * D_;
  const float* grp = gr + ((size_t)(b * NP_ + r)) * D_;
  const v4f s0 = *(const v4f*)(gsp + lane * 8);
  const v4f s1 = *(const v4f*)(gsp + lane * 8 + 4);
  const v4f r0 = *(const v4f*)(grp + lane * 8);
  const v4f r1 = *(const v4f*)(grp + lane * 8 + 4);

  asm volatile("s_wait_asynccnt 0x0" ::: "memory");
  __syncthreads();
  if (eg >= B_ * E_) return;

  float efv[16];
#pragma unroll
  for (int q = 0; q < 16; ++q) efv[q] = ef4[q >> 2][q & 3];
  float sv[8] = {s0[0], s0[1], s0[2], s0[3], s1[0], s1[1], s1[2], s1[3]};
  float rv[8] = {r0[0], r0[1], r0[2], r0[3], r1[0], r1[1], r1[2], r1[3]};

  const int head = lane >> 2;
  float lp = 0.f;
#pragma unroll
  for (int j = 0; j < 8; ++j) {
    const int d = lane * 8 + j;
    float eh = sBe[d];
#pragma unroll
    for (int q = 0; q < 16; ++q) eh += efv[q] * sWe[q * D_ + d];
    float v = sv[j] + rv[j] + eh;
    v = v > 0.f ? v : 0.2f * v;                 // leaky_relu(0.2)
    lp += v * sAt[head * DH_ + (lane & 3) * 8 + j];
  }
  lp += __shfl_xor(lp, 1, 32);
  lp += __shfl_xor(lp, 2, 32);
  if ((lane & 3) == 0) {
    const float lg = mk ? lp : NEG_;
    logits[(size_t)eg * H_ + head] = lg;
    atomicMax(&mmax[(size_t)(b * NP_ + r) * H_ + head], fenc(lg));
  }
}

// ex = exp(logit - m), atomic segment sum of ex
__global__ void gat_exp(const int* __restrict__ el, const unsigned char* __restrict__ emk,
                        const unsigned* __restrict__ mmax, float* __restrict__ logits,
                        float* __restrict__ den) {
  const int idx = blockIdx.x * 256 + threadIdx.x;
  if (idx >= B_ * E_ * H_) return;
  const int h = idx % H_;
  const int eg = idx / H_;
  const int b = eg / E_;
  const bool mk = emk[eg] != 0;
  const int r = mk ? el[(size_t)eg * 2 + 1] : N_;
  float m = fdec(mmax[(size_t)(b * NP_ + r) * H_ + h]);
  if (!(m > -3.0e38f)) m = 0.f;  // where(isfinite(m), m, 0)
  const float ex = __expf(logits[idx] - m);
  logits[idx] = ex;
  atomicAdd(&den[(size_t)(b * NP_ + r) * H_ + h], ex);
}

// msg = alpha * hs(source proj), scattered with atomics; one wave per edge
__global__ void gat_scatter(const float* __restrict__ gs, const int* __restrict__ el,
                            const unsigned char* __restrict__ emk,
                            const float* __restrict__ logits, const float* __restrict__ den,
                            float* __restrict__ outagg) {
  const int wave = threadIdx.x >> 5, lane = threadIdx.x & 31;
  const int eg = blockIdx.x * 8 + wave;
  if (eg >= B_ * E_) return;
  const int b = eg / E_;
  const bool mk = emk[eg] != 0;
  const int s = mk ? el[(size_t)eg * 2] : N_;
  const int r = mk ? el[(size_t)eg * 2 + 1] : N_;
  const int head = lane >> 2;
  const float alpha = logits[(size_t)eg * H_ + head] /
                      (den[(size_t)(b * NP_ + r) * H_ + head] + 1e-9f);
  const float* gsp = gs + ((size_t)(b * NP_ + s)) * D_;
  const v4f s0 = *(const v4f*)(gsp + lane * 8);
  const v4f s1 = *(const v4f*)(gsp + lane * 8 + 4);
  float* op = outagg + ((size_t)(b * NP_ + r)) * D_ + lane * 8;
#pragma unroll
  for (int j = 0; j < 4; ++j) atomicAdd(&op[j], alpha * s0[j]);
#pragma unroll
  for (int j = 0; j < 4; ++j) atomicAdd(&op[4 + j], alpha * s1[j]);
}

__global__ void add_relu(const float* __restrict__ a, const float* __restrict__ sk,
                         float* __restrict__ x) {
  const int i = blockIdx.x * 256 + threadIdx.x;
  if (i >= MROWS * D_) return;
  const float v = a[i] + sk[i];
  x[i] = v > 0.f ? v : 0.f;
}

// ---------------- host ----------------

struct ABp { const float *Wk,*Wo,*Wq,*Wv,*bk,*bo,*bq,*bv,*ln_b,*ln_s,*lnk,*lnq; };
struct Gp  { const float *We,*Wr,*Ws,*Wsk,*attn,*be,*br,*bs,*bsk,*ln_b,*ln_s; };

extern "C" void kernel_launch(void* const* d_in, const int* in_sizes, int n_in,
                              void* d_out, int out_size, void* d_ws, size_t ws_size,
                              hipStream_t stream) {
  (void)in_sizes; (void)out_size; (void)ws_size;
  if (n_in < 85) return;  // params pytree expected flattened (sorted-key leaf order)

  const float* node_features   = (const float*)d_in[0];
  const float* edge_features   = (const float*)d_in[1];
  const int*   edge_list       = (const int*)d_in[2];
  const unsigned char* edge_mask = (const unsigned char*)d_in[3];
  const float* global_features = (const float*)d_in[4];
  const float* node_mask       = (const float*)d_in[5];

  auto F = [&](int i) { return (const float*)d_in[i]; };
  // top-level sorted keys: W1,W2,Wc,ab0,ab1,b1,b2,bc,glob,layers,ln1_b,ln1_s,lnf_b,lnf_s
  const float* W1 = F(6); const float* W2 = F(7); const float* Wc = F(8);
  auto loadAB = [&](int base) {
    ABp a; a.Wk=F(base+0); a.Wo=F(base+1); a.Wq=F(base+2); a.Wv=F(base+3);
    a.bk=F(base+4); a.bo=F(base+5); a.bq=F(base+6); a.bv=F(base+7);
    a.ln_b=F(base+8); a.ln_s=F(base+9); a.lnk=F(base+10); a.lnq=F(base+11);
    return a;
  };
  ABp ab0 = loadAB(9), ab1 = loadAB(21);
  const float* b1 = F(33); const float* b2 = F(34); const float* bc = F(35);
  const float* glob = F(36);
  Gp Lp[4];
  for (int i = 0; i < 4; ++i) {
    const int b = 37 + i * 11;  // We,Wr,Ws,Wsk,attn,be,br,bs,bsk,ln_b,ln_s
    Lp[i].We=F(b+0); Lp[i].Wr=F(b+1); Lp[i].Ws=F(b+2); Lp[i].Wsk=F(b+3);
    Lp[i].attn=F(b+4); Lp[i].be=F(b+5); Lp[i].br=F(b+6); Lp[i].bs=F(b+7);
    Lp[i].bsk=F(b+8); Lp[i].ln_b=F(b+9); Lp[i].ln_s=F(b+10);
  }
  const float* ln1_b = F(81); const float* ln1_s = F(82);
  const float* lnf_b = F(83); const float* lnf_s = F(84);

  // workspace carve (bump allocator, 256B aligned)
  char* base = (char*)d_ws;
  size_t off = 0;
  auto alloc = [&](size_t bytes) {
    void* p = base + off;
    off += (bytes + 255) & ~(size_t)255;
    return p;
  };
  float* nf     = (float*)alloc((size_t)MROWS * 48 * 4);
  float* x      = (float*)alloc((size_t)MROWS * D_ * 4);
  float* xl     = (float*)alloc((size_t)MROWS * D_ * 4);
  float* gs     = (float*)alloc((size_t)MROWS * D_ * 4);   // note: MROWS*D_*4 is 256B-multiple,
  float* gr     = (float*)alloc((size_t)MROWS * D_ * 4);   // so gs and gr are contiguous
  float* skip   = (float*)alloc((size_t)MROWS * D_ * 4);
  float* outagg = (float*)alloc((size_t)MROWS * D_ * 4);
  float* kh     = (float*)alloc((size_t)MROWS * D_ * 4);   // also encoder h1
  float* vh     = (float*)alloc((size_t)MROWS * D_ * 4);   // also encoder h2
  float* scores = (float*)alloc((size_t)B_ * H_ * NP_ * 4);
  float* gbuf   = (float*)alloc((size_t)B_ * D_ * 4);
  float* qln    = (float*)alloc((size_t)B_ * D_ * 4);
  float* oh     = (float*)alloc((size_t)B_ * D_ * 4);
  float* logits = (float*)alloc((size_t)B_ * E_ * H_ * 4);
  unsigned* mmax = (unsigned*)alloc((size_t)MROWS * H_ * 4);
  float* den    = (float*)alloc((size_t)MROWS * H_ * 4);
  float* xc = gs;  // alias: spans gs..gr (MROWS*512 floats), used only before the layer loop

  const dim3 gGemm((MROWS + 31) / 32, D_ / 256);
  const int edgeBlocks = (B_ * E_) / 8;              // 8192
  const int expBlocks  = (B_ * E_ * H_ + 255) / 256; // 2048
  const int xBlocks    = (MROWS * D_ + 255) / 256;   // 4100

  auto run_attn = [&](const ABp& p) {
    qproj_ln<<<B_, 256, 0, stream>>>(gbuf, p.Wq, p.bq, p.lnq, qln);
    gemm_wmma<<<gGemm, 256, 0, stream>>>(x, p.Wk, p.bk, kh, MROWS, D_, D_, 0);
    gemm_wmma<<<gGemm, 256, 0, stream>>>(x, p.Wv, p.bv, vh, MROWS, D_, D_, 0);
    attn_scores<<<(B_ * NP_ * H_ + 7) / 8, 256, 0, stream>>>(kh, qln, p.lnk, node_mask, scores);
    attn_softmax_o<<<B_ * H_, 256, 0, stream>>>(scores, vh, oh);
    attn_out_ln<<<B_, 256, 0, stream>>>(oh, p.Wo, p.bo, p.ln_s, p.ln_b, gbuf);
  };

  // ---- node encoder ----
  prep_nf<<<(MROWS * 48 + 255) / 256, 256, 0, stream>>>(node_features, global_features, nf);
  gemm_wmma<<<gGemm, 256, 0, stream>>>(nf, W1, b1, kh, MROWS, 48, D_, 0);
  ln_rows<<<MROWS, 256, 0, stream>>>(kh, ln1_s, ln1_b, vh, 1);
  gemm_wmma<<<gGemm, 256, 0, stream>>>(vh, W2, b2, x, MROWS, D_, D_, 0);

  // ---- global token + first cross-attention ----
  init_g<<<(B_ * D_ + 255) / 256, 256, 0, stream>>>(glob, gbuf);
  run_attn(ab0);

  // ---- concat(x, g) @ Wc, relu ----
  concat_xg<<<(MROWS * 512 + 255) / 256, 256, 0, stream>>>(x, gbuf, xc);
  gemm_wmma<<<gGemm, 256, 0, stream>>>(xc, Wc, bc, x, MROWS, 512, D_, 1);

  // ---- GATv2 layers ----
  for (int i = 0; i < 4; ++i) {
    const Gp& q = Lp[i];
    ln_rows<<<MROWS, 256, 0, stream>>>(x, q.ln_s, q.ln_b, xl, 0);
    gemm_wmma<<<gGemm, 256, 0, stream>>>(xl, q.Ws,  q.bs,  gs,   MROWS, D_, D_, 0);
    gemm_wmma<<<gGemm, 256, 0, stream>>>(xl, q.Wr,  q.br,  gr,   MROWS, D_, D_, 0);
    gemm_wmma<<<gGemm, 256, 0, stream>>>(xl, q.Wsk, q.bsk, skip, MROWS, D_, D_, 0);
    gat_init<<<xBlocks, 256, 0, stream>>>(mmax, den, outagg);
    gat_logits<<<edgeBlocks, 256, 0, stream>>>(gs, gr, edge_features, edge_list, edge_mask,
                                               q.We, q.be, q.attn, logits, mmax);
    gat_exp<<<expBlocks, 256, 0, stream>>>(edge_list, edge_mask, mmax, logits, den);
    gat_scatter<<<edgeBlocks, 256, 0, stream>>>(gs, edge_list, edge_mask, logits, den, outagg);
    add_relu<<<xBlocks, 256, 0, stream>>>(outagg, skip, x);
  }

  // ---- second cross-attention + final LN/relu ----
  run_attn(ab1);
  final_out<<<B_, 256, 0, stream>>>(gbuf, lnf_s, lnf_b, (float*)d_out);
}